// BinaryTokenClassificationModel_32658931319384
// MI455X (gfx1250) — compile-verified
//
#include <hip/hip_runtime.h>
#include <hip/hip_bf16.h>
#include <cstdint>
#include <cstddef>

typedef float v2f __attribute__((ext_vector_type(2)));
typedef float v4f __attribute__((ext_vector_type(4)));
typedef float v8f __attribute__((ext_vector_type(8)));

constexpr int Bc = 8;       // batch
constexpr int Lc = 4096;    // sequence length
constexpr int Hc = 1024;    // hidden
constexpr int Sc = 512;     // num_src
constexpr int Tc = 512;     // num_tgt
constexpr int TOK = 64;     // tokens per block (dot kernel)
constexpr int KC  = 32;     // K chunk (floats)
constexpr int TSTRIDE = 36; // padded LDS row stride (dwords): conflict-free & 16B aligned
constexpr int WSTRIDE = 34; // padded weight-tile stride (dwords): 8B aligned, low conflict
constexpr int NSEG = 1056;  // per-batch segment-slot stride (>= 1025)

// ---------------------------------------------------------------------------
// Kernel 1: per-batch segmentation (matches reference _pool_one semantics).
// One 1024-thread block per batch, 4 tokens/thread, two block scans.
// ---------------------------------------------------------------------------
__global__ __launch_bounds__(1024) void seg_scan_kernel(
    const int* __restrict__ wid, const int* __restrict__ mask,
    int* __restrict__ seg, int* __restrict__ flag) {
  __shared__ int sHas[1024];
  __shared__ int sWid[1024];
  __shared__ int sSum[1024];
  const int tid = threadIdx.x;
  const int b = blockIdx.x;
  const int base = b * Lc + tid * 4;

  int w[4], v[4];
#pragma unroll
  for (int j = 0; j < 4; ++j) {
    w[j] = wid[base + j];
    v[j] = (mask[base + j] > 0) ? 1 : 0;
  }

  // --- scan 1: "last valid word id" prefix (for carry-in prev_wid) ---
  int curH = 0, curW = -2;
#pragma unroll
  for (int j = 0; j < 4; ++j)
    if (v[j]) { curH = 1; curW = w[j]; }
  sHas[tid] = curH; sWid[tid] = curW;
  __syncthreads();
  for (int off = 1; off < 1024; off <<= 1) {
    int pH = 0, pW = -2;
    if (tid >= off) { pH = sHas[tid - off]; pW = sWid[tid - off]; }
    __syncthreads();
    if (!curH && pH) { curH = 1; curW = pW; }   // combine: right.has ? right : left
    sHas[tid] = curH; sWid[tid] = curW;
    __syncthreads();
  }
  int prev = -2;
  if (tid > 0 && sHas[tid - 1]) prev = sWid[tid - 1];

  // --- local new-segment flags ---
  int f[4]; int localSum = 0;
#pragma unroll
  for (int j = 0; j < 4; ++j) {
    f[j] = 0;
    if (v[j]) {
      f[j] = (w[j] == -1 || prev == -1 || w[j] != prev) ? 1 : 0;
      prev = w[j];
      localSum += f[j];
    }
  }

  // --- scan 2: inclusive sum of flags -> segment ids ---
  int curS = localSum;
  sSum[tid] = curS;
  __syncthreads();
  for (int off = 1; off < 1024; off <<= 1) {
    int p = 0;
    if (tid >= off) p = sSum[tid - off];
    __syncthreads();
    curS += p;
    sSum[tid] = curS;
    __syncthreads();
  }
  int run = (tid > 0) ? sSum[tid - 1] : 0;
#pragma unroll
  for (int j = 0; j < 4; ++j) {
    int sg = -1;
    if (v[j]) { run += f[j]; sg = run - 1; }
    seg[base + j] = sg;
    flag[base + j] = f[j];
  }
}

// ---------------------------------------------------------------------------
// Kernel 2: per-token dual dot products via f32 WMMA.
// Computes D = X_tile(16xK) @ Wpad(Kx16) where Wpad cols 0/1 are w_src/w_tgt.
// Streams the 128 MiB activations once (bandwidth bound), LDS-staged.
// ---------------------------------------------------------------------------
__global__ __launch_bounds__(128) void dot_wmma_kernel(
    const float* __restrict__ X, const float* __restrict__ wvec,
    float* __restrict__ dsrc, float* __restrict__ dtgt) {
  __shared__ __align__(16) float tile[TOK * TSTRIDE];  // 64 rows x 32 K, padded
  __shared__ __align__(16) float wPad[16 * WSTRIDE];   // 16 cols x 32 K, padded

  const int tid  = threadIdx.x;
  const int lane = tid & 31;
  const int wave = tid >> 5;              // 4 waves, 16 tokens each
  const int col  = lane & 15;             // M for A-frag, N for B-frag
  const int half = lane >> 4;             // K sub-offset selector
  const int tokBase = blockIdx.x * TOK;

  // zero the padded weight tile once (cols 2..15 stay zero forever)
  for (int i = tid; i < 16 * WSTRIDE; i += 128) wPad[i] = 0.0f;

  v8f acc = {};

  for (int k0 = 0; k0 < Hc; k0 += KC) {
    __syncthreads();  // previous chunk's LDS reads done

    // refill weight columns 0 (w_src) and 1 (w_tgt) for this K chunk
    if (tid < 64) {
      int r = tid >> 5, c = tid & 31;
      wPad[r * WSTRIDE + c] = wvec[r * Hc + k0 + c];
    }
    // coalesced fill of 64x32 activation tile (float4 per thread, 4 iters)
#pragma unroll
    for (int it = 0; it < 4; ++it) {
      int idx = it * 128 + tid;           // 0..511
      int row = idx >> 3, c4 = idx & 7;
      v4f val = *reinterpret_cast<const v4f*>(
          X + (size_t)(tokBase + row) * Hc + k0 + c4 * 4);
      *reinterpret_cast<v4f*>(&tile[row * TSTRIDE + c4 * 4]) = val;
    }
    // prefetch next K chunk (one cacheline per row) -> global_prefetch_b8
    if (k0 + KC < Hc && tid < TOK) {
      __builtin_prefetch(X + (size_t)(tokBase + tid) * Hc + k0 + KC, 0, 1);
    }
    __syncthreads();

    const float* aBase = &tile[(wave * 16 + col) * TSTRIDE + half * 2];
    const float* bBase = &wPad[col * WSTRIDE + half * 2];
#pragma unroll
    for (int s = 0; s < 8; ++s) {         // 8 x (16x16x4) covers K chunk of 32
      v2f a = *reinterpret_cast<const v2f*>(aBase + s * 4);
      v2f bf = *reinterpret_cast<const v2f*>(bBase + s * 4);
      acc = __builtin_amdgcn_wmma_f32_16x16x4_f32(
          /*neg_a=*/false, a, /*neg_b=*/false, bf,
          /*c_mod=*/(short)0, acc, /*reuse_a=*/false, /*reuse_b=*/false);
    }
  }

  // C/D layout: VGPR r, lanes 0-15 -> (M=r, N=lane); lanes 16-31 -> (M=8+r).
  // Column N=0 holds src dots, N=1 holds tgt dots.
  if (col < 2) {
    float* dst = col ? dtgt : dsrc;
    int rb = tokBase + wave * 16 + half * 8;
#pragma unroll
    for (int r = 0; r < 8; ++r) dst[rb + r] = acc[r];
  }
}

// ---------------------------------------------------------------------------
// Kernel 3: deterministic segment reduction (no atomics).
// The thread at each segment start walks its contiguous run.
// ---------------------------------------------------------------------------
__global__ void seg_reduce_kernel(
    const int* __restrict__ seg, const int* __restrict__ flag,
    const float* __restrict__ dsrc, const float* __restrict__ dtgt,
    float* __restrict__ sums, int* __restrict__ counts) {
  int idx = blockIdx.x * blockDim.x + threadIdx.x;
  if (idx >= Bc * Lc) return;
  if (!flag[idx]) return;
  int sg = seg[idx];
  if (sg < 1 || sg > Sc + Tc) return;     // only segments feeding src/tgt
  int b = idx / Lc;
  int end = (b + 1) * Lc;
  const float* d = (sg <= Sc) ? dsrc : dtgt;
  float sum = 0.0f; int cnt = 0;
  for (int j = idx; j < end; ++j) {
    int s2 = seg[j];
    if (s2 == sg) { sum += d[j]; ++cnt; }
    else if (s2 != -1) break;             // next segment started
  }
  sums[b * NSEG + sg] = sum;
  counts[b * NSEG + sg] = cnt;
}

// ---------------------------------------------------------------------------
// Kernel 4: per-segment means -> a[b,s], c[b,t]
// ---------------------------------------------------------------------------
__global__ void avg_kernel(const float* __restrict__ sums,
                           const int* __restrict__ counts,
                           float* __restrict__ avg_a, float* __restrict__ avg_c) {
  int idx = blockIdx.x * blockDim.x + threadIdx.x;
  if (idx >= Bc * (Sc + Tc)) return;
  int b = idx / (Sc + Tc);
  int i = idx % (Sc + Tc);
  int sg = 1 + i;                          // src: 1..Sc ; tgt: Sc+1..Sc+Tc
  float cnt = (float)max(counts[b * NSEG + sg], 1);
  float m = sums[b * NSEG + sg] / cnt;
  if (i < Sc) avg_a[b * Sc + i] = m;
  else        avg_c[b * Tc + (i - Sc)] = m;
}

// ---------------------------------------------------------------------------
// Kernel 5: logits[b,s,t] = a[b,s] + c[b,t] + bias   (float4 stores)
// ---------------------------------------------------------------------------
__global__ void logits_kernel(const float* __restrict__ a,
                              const float* __restrict__ c,
                              const float* __restrict__ biasPtr,
                              float* __restrict__ out) {
  int idx = blockIdx.x * blockDim.x + threadIdx.x;   // over B*S*T/4
  if (idx >= Bc * Sc * (Tc / 4)) return;
  int t4  = idx % (Tc / 4);
  int rem = idx / (Tc / 4);
  int s = rem % Sc;
  int b = rem / Sc;
  float av = a[b * Sc + s] + biasPtr[0];
  v4f cv = *reinterpret_cast<const v4f*>(c + b * Tc + t4 * 4);
  v4f o = cv + av;
  reinterpret_cast<v4f*>(out)[idx] = o;
}

// ---------------------------------------------------------------------------
extern "C" void kernel_launch(void* const* d_in, const int* in_sizes, int n_in,
                              void* d_out, int out_size, void* d_ws, size_t ws_size,
                              hipStream_t stream) {
  const float* X    = (const float*)d_in[0];   // (B,L,H) f32
  const int*   msk  = (const int*)d_in[1];     // (B,L) i32
  const int*   wid  = (const int*)d_in[2];     // (B,L) i32
  // d_in[3]/d_in[4]: num_src/num_tgt == 512 (module constants, hardcoded)
  const float* wvec = (const float*)d_in[5];   // (2H,) f32
  const float* bias = (const float*)d_in[6];   // scalar f32
  float* out = (float*)d_out;

  // workspace layout
  char* ws = (char*)d_ws;
  size_t off = 0;
  int*   seg    = (int*)(ws + off);   off += (size_t)Bc * Lc * 4;
  int*   flg    = (int*)(ws + off);   off += (size_t)Bc * Lc * 4;
  float* dsrc   = (float*)(ws + off); off += (size_t)Bc * Lc * 4;
  float* dtgt   = (float*)(ws + off); off += (size_t)Bc * Lc * 4;
  char*  zbase  = ws + off;
  float* sums   = (float*)(ws + off); off += (size_t)Bc * NSEG * 4;
  int*   counts = (int*)(ws + off);   off += (size_t)Bc * NSEG * 4;
  size_t zbytes = (size_t)Bc * NSEG * 8;
  float* avg_a  = (float*)(ws + off); off += (size_t)Bc * Sc * 4;
  float* avg_c  = (float*)(ws + off); off += (size_t)Bc * Tc * 4;

  // zero the reduction targets every launch (graph-capture safe)
  hipMemsetAsync(zbase, 0, zbytes, stream);

  // 1) segmentation
  seg_scan_kernel<<<Bc, 1024, 0, stream>>>(wid, msk, seg, flg);

  // 2) per-token dual dot products via WMMA (streams 128 MiB once)
  dot_wmma_kernel<<<(Bc * Lc) / TOK, 128, 0, stream>>>(X, wvec, dsrc, dtgt);

  // 3) deterministic segment sums/counts
  seg_reduce_kernel<<<(Bc * Lc + 255) / 256, 256, 0, stream>>>(
      seg, flg, dsrc, dtgt, sums, counts);

  // 4) means
  avg_kernel<<<(Bc * (Sc + Tc) + 255) / 256, 256, 0, stream>>>(
      sums, counts, avg_a, avg_c);

  // 5) outer-sum logits
  logits_kernel<<<(Bc * Sc * (Tc / 4) + 255) / 256, 256, 0, stream>>>(
      avg_a, avg_c, bias, out);
}